// AttnDecoderRNN_51462298140791
// MI455X (gfx1250) — compile-verified
//
#include <hip/hip_runtime.h>
#include <math.h>
#include <float.h>

// Problem dimensions (match reference)
#define TOKV 50000   // vocab
#define EDIM 1024    // embedding dim
#define HDIM 1024    // hidden dim
#define SLEN 4096    // encoder sequence length
#define KCAT 2048    // 2*H (fc input)

typedef __attribute__((ext_vector_type(2))) float v2f;
typedef __attribute__((ext_vector_type(8))) float v8f;

// ---------------------------------------------------------------------------
// Wave-level GEMV via V_WMMA_F32_16X16X4_F32.
// One wave computes 16 output rows: D = A x B with
//   A[m][k] = x[k0+k]  (vector broadcast across M)
//   B[k][n] = W[n_base+n][k0+k]
// A-layout (ISA 7.12.2, 32-bit A 16x4): VGPR0 = K0 (lanes 0-15) / K2 (16-31),
// VGPR1 = K1 / K3.  B mirrors the K split; row n striped across lanes.
// Result: logits for n = 0..15 are acc[0] on lanes 0..15 (M=0 row of D).
// Lanes l and l+16 stream row l together (8B each per chunk) -> full lines.
// ---------------------------------------------------------------------------
__device__ __forceinline__ v8f wmma_gemv16(const float* __restrict__ wrow,
                                           const float* __restrict__ xvec,
                                           int K, int lane, v8f acc)
{
  const int ko = (lane < 16) ? 0 : 2;
#pragma unroll 4
  for (int k0 = 0; k0 < K; k0 += 4) {
    const float2 wv = *(const float2*)(wrow + k0 + ko);
    const float2 av = *(const float2*)(xvec + k0 + ko);
    v2f a; a.x = av.x; a.y = av.y;
    v2f b; b.x = wv.x; b.y = wv.y;
    acc = __builtin_amdgcn_wmma_f32_16x16x4_f32(false, a, false, b,
                                                (short)0, acc, false, false);
  }
  return acc;
}

__device__ __forceinline__ float block_max_256(float v, float* red) {
  int t = threadIdx.x;
  red[t] = v; __syncthreads();
  for (int s = 128; s > 0; s >>= 1) {
    if (t < s) red[t] = fmaxf(red[t], red[t + s]);
    __syncthreads();
  }
  float r = red[0]; __syncthreads();
  return r;
}

__device__ __forceinline__ float block_sum_256(float v, float* red) {
  int t = threadIdx.x;
  red[t] = v; __syncthreads();
  for (int s = 128; s > 0; s >>= 1) {
    if (t < s) red[t] += red[t + s];
    __syncthreads();
  }
  float r = red[0]; __syncthreads();
  return r;
}

// ---------------------------------------------------------------------------
// K1: gates[4096] = w_ih @ x + w_hh @ h  (biases added at store)
// 32 blocks x 256 threads = 256 waves, 16 gate rows per wave.
// ---------------------------------------------------------------------------
__global__ __launch_bounds__(256)
void k_gates(const int* __restrict__ tok, const float* __restrict__ emb,
             const float* __restrict__ h0,
             const float* __restrict__ w_ih, const float* __restrict__ w_hh,
             const float* __restrict__ b_ih, const float* __restrict__ b_hh,
             float* __restrict__ gates)
{
  __shared__ float xs[EDIM];
  __shared__ float hs[HDIM];
  const int t = threadIdx.x;
  const float* x = emb + (size_t)tok[0] * EDIM;
  for (int i = t; i < EDIM; i += 256) xs[i] = x[i];
  for (int i = t; i < HDIM; i += 256) hs[i] = h0[i];
  __syncthreads();

  const int wave   = (blockIdx.x * 256 + t) >> 5;
  const int lane   = t & 31;
  const int n_base = wave * 16;
  const int row    = n_base + (lane & 15);

  v8f acc = {};
  acc = wmma_gemv16(w_ih + (size_t)row * EDIM, xs, EDIM, lane, acc);
  acc = wmma_gemv16(w_hh + (size_t)row * HDIM, hs, HDIM, lane, acc);

  if (lane < 16) {
    const int n = n_base + lane;
    gates[n] = acc[0] + b_ih[n] + b_hh[n];
  }
}

// ---------------------------------------------------------------------------
// K2: LSTM pointwise: i,f,g,o -> cn, hn.  Writes ws copies and d_out tail.
// ---------------------------------------------------------------------------
__global__ __launch_bounds__(256)
void k_lstm(const float* __restrict__ gates, const float* __restrict__ c0,
            float* __restrict__ hn, float* __restrict__ cn,
            float* __restrict__ out_hn, float* __restrict__ out_cn)
{
  const int j = blockIdx.x * 256 + threadIdx.x;
  if (j >= HDIM) return;
  const float gi = gates[j];
  const float gf = gates[j + HDIM];
  const float gg = gates[j + 2 * HDIM];
  const float go = gates[j + 3 * HDIM];
  const float i = 1.0f / (1.0f + expf(-gi));
  const float f = 1.0f / (1.0f + expf(-gf));
  const float g = tanhf(gg);
  const float o = 1.0f / (1.0f + expf(-go));
  const float c = f * c0[j] + i * g;
  const float h = o * tanhf(c);
  cn[j] = c; hn[j] = h;
  out_cn[j] = c; out_hn[j] = h;
}

// ---------------------------------------------------------------------------
// K3: scores[4096] = enc @ hn   (WMMA GEMV, 32 blocks x 8 waves)
// ---------------------------------------------------------------------------
__global__ __launch_bounds__(256)
void k_scores(const float* __restrict__ enc, const float* __restrict__ hn,
              float* __restrict__ scores)
{
  __shared__ float hs[HDIM];
  const int t = threadIdx.x;
  for (int i = t; i < HDIM; i += 256) hs[i] = hn[i];
  __syncthreads();

  const int wave   = (blockIdx.x * 256 + t) >> 5;
  const int lane   = t & 31;
  const int n_base = wave * 16;
  const int row    = n_base + (lane & 15);

  v8f acc = {};
  acc = wmma_gemv16(enc + (size_t)row * HDIM, hs, HDIM, lane, acc);

  if (lane < 16) scores[n_base + lane] = acc[0];
}

// ---------------------------------------------------------------------------
// K4: in-place softmax over scores[4096], single block.
// ---------------------------------------------------------------------------
__global__ __launch_bounds__(256)
void k_softmax4096(float* __restrict__ s)
{
  __shared__ float red[256];
  const int t = threadIdx.x;
  float m = -FLT_MAX;
  for (int i = t; i < SLEN; i += 256) m = fmaxf(m, s[i]);
  m = block_max_256(m, red);
  float sum = 0.0f;
  for (int i = t; i < SLEN; i += 256) sum += expf(s[i] - m);
  sum = block_sum_256(sum, red);
  const float inv = 1.0f / sum;
  for (int i = t; i < SLEN; i += 256) s[i] = expf(s[i] - m) * inv;
}

// ---------------------------------------------------------------------------
// K5: context partials: 32 blocks, block b reduces rows [128b, 128b+128)
// into part[b][0..1023]; thread t owns 4 consecutive cols (float4 coalesced).
// ---------------------------------------------------------------------------
__global__ __launch_bounds__(256)
void k_ctx_partial(const float* __restrict__ enc, const float* __restrict__ p,
                   float* __restrict__ part)
{
  const int b = blockIdx.x, t = threadIdx.x;
  float4 acc = make_float4(0.f, 0.f, 0.f, 0.f);
  const int r0 = b * 128;
  for (int r = r0; r < r0 + 128; ++r) {
    const float pr = p[r];
    const float4 e = *(const float4*)(enc + (size_t)r * HDIM + t * 4);
    acc.x += pr * e.x; acc.y += pr * e.y;
    acc.z += pr * e.z; acc.w += pr * e.w;
  }
  *(float4*)(part + (size_t)b * HDIM + t * 4) = acc;
}

// ---------------------------------------------------------------------------
// K6: reduce 32 partials -> context; build concat = [hn, context]
// ---------------------------------------------------------------------------
__global__ __launch_bounds__(256)
void k_ctx_concat(const float* __restrict__ part, const float* __restrict__ hn,
                  float* __restrict__ concat)
{
  const int j = blockIdx.x * 256 + threadIdx.x;
  if (j >= HDIM) return;
  float c = 0.0f;
#pragma unroll
  for (int b = 0; b < 32; ++b) c += part[(size_t)b * HDIM + j];
  concat[j] = hn[j];
  concat[HDIM + j] = c;
}

// ---------------------------------------------------------------------------
// K7: logits[50000] = fc_w @ concat + fc_b  (WMMA GEMV, dominant 410 MB read)
// 391 blocks x 8 waves = 3128 waves; 3125 needed; tail waves clamp rows so
// EXEC stays all-ones for WMMA, and skip the store.
// ---------------------------------------------------------------------------
__global__ __launch_bounds__(256)
void k_logits(const float* __restrict__ fc_w, const float* __restrict__ fc_b,
              const float* __restrict__ concat, float* __restrict__ logits)
{
  __shared__ float cs[KCAT];
  const int t = threadIdx.x;
  for (int i = t; i < KCAT; i += 256) cs[i] = concat[i];
  __syncthreads();

  const int wave   = (blockIdx.x * 256 + t) >> 5;
  const int lane   = t & 31;
  const int n_base = wave * 16;
  const bool valid = (n_base < TOKV);
  const int  nb    = valid ? n_base : (TOKV - 16);
  const int  row   = nb + (lane & 15);

  v8f acc = {};
  acc = wmma_gemv16(fc_w + (size_t)row * KCAT, cs, KCAT, lane, acc);

  if (valid && lane < 16) {
    const int n = n_base + lane;
    logits[n] = acc[0] + fc_b[n];
  }
}

// ---------------------------------------------------------------------------
// K8/K9/K10: block-wise log-sum-exp over 50000 logits, then subtract.
// ---------------------------------------------------------------------------
#define NB_LSE 196  // ceil(50000/256)

__global__ __launch_bounds__(256)
void k_lse_partial(const float* __restrict__ logits,
                   float* __restrict__ pm, float* __restrict__ ps)
{
  __shared__ float red[256];
  const int i = blockIdx.x * 256 + threadIdx.x;
  const float x = (i < TOKV) ? logits[i] : -FLT_MAX;
  const float m = block_max_256(x, red);
  const float e = (i < TOKV) ? expf(x - m) : 0.0f;
  const float s = block_sum_256(e, red);
  if (threadIdx.x == 0) { pm[blockIdx.x] = m; ps[blockIdx.x] = s; }
}

__global__ __launch_bounds__(256)
void k_lse_combine(const float* __restrict__ pm, const float* __restrict__ ps,
                   float* __restrict__ logz)
{
  __shared__ float red[256];
  const int t = threadIdx.x;
  float m = (t < NB_LSE) ? pm[t] : -FLT_MAX;
  const float M = block_max_256(m, red);
  float s = (t < NB_LSE) ? ps[t] * expf(pm[t] - M) : 0.0f;
  const float S = block_sum_256(s, red);
  if (t == 0) logz[0] = M + logf(S);
}

__global__ __launch_bounds__(256)
void k_logp(const float* __restrict__ logits, const float* __restrict__ logz,
            float* __restrict__ out)
{
  const int i = blockIdx.x * 256 + threadIdx.x;
  if (i < TOKV) out[i] = logits[i] - logz[0];
}

// ---------------------------------------------------------------------------
// Workspace layout (float offsets)
// ---------------------------------------------------------------------------
enum : size_t {
  WS_GATES   = 0,                      // 4096
  WS_HN      = WS_GATES + 4096,        // 1024
  WS_CN      = WS_HN + 1024,           // 1024
  WS_SCORES  = WS_CN + 1024,           // 4096 (scores -> probs in place)
  WS_CONCAT  = WS_SCORES + 4096,       // 2048
  WS_CTXPART = WS_CONCAT + 2048,       // 32*1024
  WS_LOGITS  = WS_CTXPART + 32 * 1024, // 50000
  WS_PM      = WS_LOGITS + 50000,      // 196
  WS_PS      = WS_PM + NB_LSE,         // 196
  WS_LOGZ    = WS_PS + NB_LSE          // 1
};

extern "C" void kernel_launch(void* const* d_in, const int* in_sizes, int n_in,
                              void* d_out, int out_size, void* d_ws, size_t ws_size,
                              hipStream_t stream)
{
  (void)in_sizes; (void)n_in; (void)out_size; (void)ws_size;
  const int*   tok   = (const int*)d_in[0];
  const float* h0    = (const float*)d_in[1];
  const float* c0    = (const float*)d_in[2];
  const float* enc   = (const float*)d_in[3];
  const float* emb   = (const float*)d_in[4];
  const float* w_ih  = (const float*)d_in[5];
  const float* w_hh  = (const float*)d_in[6];
  const float* b_ih  = (const float*)d_in[7];
  const float* b_hh  = (const float*)d_in[8];
  const float* fc_w  = (const float*)d_in[9];
  const float* fc_b  = (const float*)d_in[10];
  float* out = (float*)d_out;      // [logp(50000) | hn(1024) | cn(1024)]
  float* ws  = (float*)d_ws;

  k_gates<<<32, 256, 0, stream>>>(tok, emb, h0, w_ih, w_hh, b_ih, b_hh,
                                  ws + WS_GATES);
  k_lstm<<<4, 256, 0, stream>>>(ws + WS_GATES, c0, ws + WS_HN, ws + WS_CN,
                                out + TOKV, out + TOKV + HDIM);
  k_scores<<<32, 256, 0, stream>>>(enc, ws + WS_HN, ws + WS_SCORES);
  k_softmax4096<<<1, 256, 0, stream>>>(ws + WS_SCORES);
  k_ctx_partial<<<32, 256, 0, stream>>>(enc, ws + WS_SCORES, ws + WS_CTXPART);
  k_ctx_concat<<<4, 256, 0, stream>>>(ws + WS_CTXPART, ws + WS_HN,
                                      ws + WS_CONCAT);
  k_logits<<<391, 256, 0, stream>>>(fc_w, fc_b, ws + WS_CONCAT,
                                    ws + WS_LOGITS);
  k_lse_partial<<<NB_LSE, 256, 0, stream>>>(ws + WS_LOGITS, ws + WS_PM,
                                            ws + WS_PS);
  k_lse_combine<<<1, 256, 0, stream>>>(ws + WS_PM, ws + WS_PS, ws + WS_LOGZ);
  k_logp<<<NB_LSE, 256, 0, stream>>>(ws + WS_LOGITS, ws + WS_LOGZ, out);
}